// WordBertNet_12163347383020
// MI455X (gfx1250) — compile-verified
//
#include <hip/hip_runtime.h>

// Problem constants from the reference: B=64, S=1024, H=768, W=128.
#define B_ 64
#define S_ 1024
#define H_ 768
#define W_ 128

typedef __attribute__((ext_vector_type(2))) float v2f;
typedef __attribute__((ext_vector_type(8))) float v8f;

// ---------------------------------------------------------------------------
// Kernel 1: trivial pass-through of rep_text into the head of d_out.
// ---------------------------------------------------------------------------
__global__ void copy_rep_text_kernel(const float* __restrict__ rep_text,
                                     float* __restrict__ out) {
    int i = blockIdx.x * blockDim.x + threadIdx.x;
    if (i < B_ * H_) out[i] = rep_text[i];
}

// ---------------------------------------------------------------------------
// Kernel 2: word mean-pool as 0/1-weighted f32 WMMA.
//
// One workgroup = (sample b, group of 16 words). 8 waves; each wave computes
// 6 of the 48 [16 words x 16 cols] output tiles:
//   tile C[16x16] = sum_{wc=0..15} A_wc[16x4] * B_wc[4x16]   (V_WMMA_F32_16X16X4_F32)
// A_wc is the one-hot/len-mask for word wc (tile-invariant, hoisted); B_wc is
// rows st_wc..st_wc+3 of this 16-col H slab, loaded UNCONDITIONALLY: rows past
// len_wc belong to the next word (spans partition consecutive rows), so they
// are valid in-bounds data that the A weights zero out exactly — no divergent
// predication around loads, and no extra HBM traffic (those lines are read by
// the neighboring word anyway / hit in cache). Mean is a per-word reciprocal
// (one exact IEEE divide per word per WG) applied as v_mul in the store loop.
//
// VGPR layouts (wave32, CDNA5 ISA):
//   A 16x4 : M = lane&15 ; lanes 0-15 hold K=0(v0),K=1(v1); lanes 16-31 K=2,K=3
//   B 4x16 : N = lane&15 ; lanes 0-15 hold K=0(v0),K=1(v1); lanes 16-31 K=2,K=3
//   C 16x16: VGPR r holds M=r (lanes 0-15) / M=r+8 (lanes 16-31), N = lane&15
// ---------------------------------------------------------------------------
__global__ __launch_bounds__(256)
void word_pool_wmma_kernel(const float* __restrict__ sub,   // [B,S,H]
                           const int*   __restrict__ lens,  // [B,W]
                           float* __restrict__ pooled) {    // [B,W,H]
    const int b  = blockIdx.x >> 3;   // sample
    const int g  = blockIdx.x & 7;    // 16-word group within sample
    const int w0 = g * 16;

    __shared__ int   s_start[16];
    __shared__ int   s_len[16];
    __shared__ float s_rcp[16];       // 1/len per word (computed once)

    const int tid = threadIdx.x;
    if (tid < 16) {
        const int* l = lens + b * W_;
        int st = 0;
        for (int k = 0; k < w0 + tid; ++k) st += l[k];  // csum[w-1]
        const int len = l[w0 + tid];
        s_start[tid] = st;
        s_len[tid]   = len;
        s_rcp[tid]   = 1.0f / (float)len;  // exact for len in {1,2,4}
    }
    __syncthreads();

    const int wave = tid >> 5;        // 8 waves of 32 (wave32)
    const int lane = tid & 31;
    const int col  = lane & 15;       // N for B/C, M for A
    const int k0   = (lane >> 4) * 2; // K of first VGPR in this lane half

    // A fragments: tile-invariant one-hot/len masks, built once.
    v2f afrag[16];
    #pragma unroll
    for (int wc = 0; wc < 16; ++wc) {
        const int len = s_len[wc];
        afrag[wc][0] = ((col == wc) && (k0     < len)) ? 1.0f : 0.0f;
        afrag[wc][1] = ((col == wc) && (k0 + 1 < len)) ? 1.0f : 0.0f;
    }

    // Subword rows with CLS (row 0) dropped: row p -> sub[b, 1+p, :].
    // Per-lane base points at (row k0, column h0+col) for h0 = 0.
    const float* vec   = sub + (size_t)b * S_ * H_ + H_ + (size_t)k0 * H_ + col;
    float*       obase = pooled + ((size_t)b * W_ + w0) * H_;

    for (int tile = wave; tile < H_ / 16; tile += 8) {
        const int h0 = tile * 16;

        // Gather block: 32 unconditional, clause-able loads.
        float bv0[16], bv1[16];
        #pragma unroll
        for (int wc = 0; wc < 16; ++wc) {
            const float* p = vec + (size_t)s_start[wc] * H_ + h0;
            bv0[wc] = p[0];    // row st+k0
            bv1[wc] = p[H_];   // row st+k0+1
        }

        // 16 chained f32 WMMAs accumulate the 16x16 tile.
        v8f c = {};
        #pragma unroll
        for (int wc = 0; wc < 16; ++wc) {
            v2f bb = {bv0[wc], bv1[wc]};
            c = __builtin_amdgcn_wmma_f32_16x16x4_f32(
                    /*neg_a=*/false, afrag[wc], /*neg_b=*/false, bb,
                    /*c_mod=*/(short)0, c, /*reuse_a=*/false, /*reuse_b=*/false);
        }

        // Mean via precomputed reciprocal: one v_mul per element.
        const int mofs = (lane >> 4) * 8;
        #pragma unroll
        for (int r = 0; r < 8; ++r) {
            const int m = r + mofs;
            obase[(size_t)m * H_ + h0 + col] = c[r] * s_rcp[m];
        }
    }
}

// ---------------------------------------------------------------------------
// Host-side launcher (graph-capture safe: only kernel launches on `stream`).
// Inputs: d_in[0]=rep_text f32[B,H], d_in[1]=rep_sub_words f32[B,S,H],
//         d_in[2]=list_lens i32[B,W].
// Output: d_out = [rep_text | pooled] flat f32.
// ---------------------------------------------------------------------------
extern "C" void kernel_launch(void* const* d_in, const int* in_sizes, int n_in,
                              void* d_out, int out_size, void* d_ws, size_t ws_size,
                              hipStream_t stream) {
    (void)in_sizes; (void)n_in; (void)out_size; (void)d_ws; (void)ws_size;

    const float* rep_text = (const float*)d_in[0];
    const float* sub      = (const float*)d_in[1];
    const int*   lens     = (const int*)d_in[2];

    float* out_text   = (float*)d_out;            // first B*H floats
    float* out_pooled = out_text + B_ * H_;       // then B*W*H floats

    copy_rep_text_kernel<<<(B_ * H_ + 255) / 256, 256, 0, stream>>>(rep_text, out_text);
    word_pool_wmma_kernel<<<B_ * 8, 256, 0, stream>>>(sub, lens, out_pooled);
}